// GATLayer_62148176773134
// MI455X (gfx1250) — compile-verified
//
#include <hip/hip_runtime.h>
#include <math.h>

// ---------------------------------------------------------------------------
// 2-layer GAT on MI455X (gfx1250, wave32).
//   - Edge softmax/aggregation: memory/atomic bound, ~410MB gather per layer
//     -> plain float atomics into L2-resident 25.6MB accumulators.
//   - h @ W2 (50000x128 @ 128x128 fp32): V_WMMA_F32_16X16X4_F32, one 16x16
//     output tile per wave, A-stripe staged in LDS, 32 WMMA per tile.
//     (Verified: compiler pipelines ds_load_b64 + coalesced global_load_b32
//      against the WMMA chain; prefetch removed — it lowered to SYS-scope
//      global_prefetch_b8 which duplicates the demand loads.)
// ---------------------------------------------------------------------------

#define N_NODES 50000
#define N_EDGES 800000
#define HID 128

typedef __attribute__((ext_vector_type(2))) float v2f;
typedef __attribute__((ext_vector_type(8))) float v8f;

__device__ __forceinline__ float waveSum(float v) {
#pragma unroll
  for (int off = 16; off > 0; off >>= 1) v += __shfl_xor(v, off, 32);
  return v;
}

__device__ __forceinline__ void atomicMaxF32(float* addr, float val) {
  // IEEE ordering trick: signed-int max for >=0, unsigned min for <0.
  if (val >= 0.0f) atomicMax((int*)addr, __float_as_int(val));
  else             atomicMin((unsigned int*)addr, __float_as_uint(val));
}

// ---- Layer 1: z = x @ W1 (K=2), el = z.al, er = z.ar. One wave per node. ----
__global__ void __launch_bounds__(256)
k_l1_node(const float* __restrict__ x, const float* __restrict__ W1,
          const float* __restrict__ al, const float* __restrict__ ar,
          float* __restrict__ Z, float* __restrict__ el, float* __restrict__ er) {
  int wid  = (blockIdx.x * blockDim.x + threadIdx.x) >> 5;
  int lane = threadIdx.x & 31;
  if (wid >= N_NODES) return;
  float x0 = x[wid * 2 + 0];
  float x1 = x[wid * 2 + 1];
  int c = lane * 4;
  float4 z;
  float pel = 0.0f, per = 0.0f;
#pragma unroll
  for (int i = 0; i < 4; ++i) {
    int j = c + i;
    float zj = x0 * W1[j] + x1 * W1[HID + j];
    ((float*)&z)[i] = zj;
    pel += zj * al[j];
    per += zj * ar[j];
  }
  *(float4*)&Z[wid * HID + c] = z;
  pel = waveSum(pel);
  per = waveSum(per);
  if (lane == 0) { el[wid] = pel; er[wid] = per; }
}

// ---- zero aggregation target, init segment max/denom ----
__global__ void __launch_bounds__(256)
k_init(float* __restrict__ H, float* __restrict__ m, float* __restrict__ denom) {
  int tid = blockIdx.x * blockDim.x + threadIdx.x;
  if (tid < N_NODES * HID) H[tid] = 0.0f;
  if (tid < N_NODES) { m[tid] = -__builtin_inff(); denom[tid] = 0.0f; }
}

// ---- edge logits + per-dst max ----
__global__ void __launch_bounds__(256)
k_edge_logits(const int* __restrict__ src, const int* __restrict__ dst,
              const float* __restrict__ el, const float* __restrict__ er,
              float* __restrict__ ex, float* __restrict__ m) {
  int e = blockIdx.x * blockDim.x + threadIdx.x;
  if (e >= N_EDGES) return;
  float v = el[src[e]] + er[dst[e]];
  v = (v > 0.0f) ? v : 0.2f * v;          // leaky_relu 0.2
  ex[e] = v;
  atomicMaxF32(&m[dst[e]], v);
}

// ---- exp(e - m[dst]) + per-dst denom ----
__global__ void __launch_bounds__(256)
k_edge_exp(const int* __restrict__ dst, const float* __restrict__ m,
           float* __restrict__ ex, float* __restrict__ denom) {
  int e = blockIdx.x * blockDim.x + threadIdx.x;
  if (e >= N_EDGES) return;
  int d = dst[e];
  float v = expf(ex[e] - m[d]);
  ex[e] = v;
  atomicAdd(&denom[d], v);
}

// ---- alpha-weighted scatter: one wave per edge, float4 per lane ----
__global__ void __launch_bounds__(256)
k_aggregate(const int* __restrict__ src, const int* __restrict__ dst,
            const float* __restrict__ ex, const float* __restrict__ denom,
            const float* __restrict__ Z, float* __restrict__ H) {
  int wid  = (blockIdx.x * blockDim.x + threadIdx.x) >> 5;
  int lane = threadIdx.x & 31;
  if (wid >= N_EDGES) return;
  int s = src[wid], d = dst[wid];
  float w = ex[wid] / (denom[d] + 1e-9f);
  int c = lane * 4;
  float4 z = *(const float4*)&Z[s * HID + c];
  atomicAdd(&H[d * HID + c + 0], w * z.x);
  atomicAdd(&H[d * HID + c + 1], w * z.y);
  atomicAdd(&H[d * HID + c + 2], w * z.z);
  atomicAdd(&H[d * HID + c + 3], w * z.w);
}

// ---- bias + relu, elementwise over N*HID ----
__global__ void __launch_bounds__(256)
k_bias_relu(float* __restrict__ H, const float* __restrict__ b) {
  int tid = blockIdx.x * blockDim.x + threadIdx.x;
  if (tid >= N_NODES * HID) return;
  float v = H[tid] + b[tid & (HID - 1)];
  H[tid] = (v > 0.0f) ? v : 0.0f;
}

// ---- D = A(16x128 stripe) @ B(128x128) via V_WMMA_F32_16X16X4_F32 ----
// One block per 16-row stripe; 8 waves cover the 8 column tiles.
__global__ void __launch_bounds__(256)
k_gemm_wmma(const float* __restrict__ A, const float* __restrict__ B,
            float* __restrict__ D) {
  __shared__ float As[16 * 132];          // 132-stride pad vs bank conflicts
  int rowBase = blockIdx.x * 16;
  int tid = threadIdx.x;
#pragma unroll
  for (int i = tid; i < 16 * HID; i += 256) {
    int r = i >> 7, c = i & 127;
    As[r * 132 + c] = A[(rowBase + r) * HID + c];
  }
  __syncthreads();

  int wave = tid >> 5, lane = tid & 31;
  int colBase = wave * 16;
  int hi   = lane >> 4;                   // half-wave select
  int ln   = lane & 15;
  int koff = hi * 2;                      // lanes 0-15: K={0,1}; 16-31: K={2,3}
  v8f acc = {};
  for (int k0 = 0; k0 < HID; k0 += 4) {
    v2f a, b;
    a.x = As[ln * 132 + k0 + koff];       // A: M=ln, K=k0+koff{,+1}
    a.y = As[ln * 132 + k0 + koff + 1];
    b.x = B[(k0 + koff) * HID + colBase + ln];      // B: K rows, N=colBase+ln
    b.y = B[(k0 + koff + 1) * HID + colBase + ln];
    acc = __builtin_amdgcn_wmma_f32_16x16x4_f32(false, a, false, b,
                                                (short)0, acc, false, false);
  }
  // C/D layout: VGPR r -> row r (lanes 0-15) / row r+8 (lanes 16-31), col=ln
#pragma unroll
  for (int r = 0; r < 8; ++r) {
    D[(rowBase + r + hi * 8) * HID + colBase + ln] = acc[r];
  }
}

// ---- el/er for layer 2: wave per node ----
__global__ void __launch_bounds__(256)
k_node_attn(const float* __restrict__ Z, const float* __restrict__ al,
            const float* __restrict__ ar, float* __restrict__ el,
            float* __restrict__ er) {
  int wid  = (blockIdx.x * blockDim.x + threadIdx.x) >> 5;
  int lane = threadIdx.x & 31;
  if (wid >= N_NODES) return;
  int c = lane * 4;
  float4 z = *(const float4*)&Z[wid * HID + c];
  float pel = 0.0f, per = 0.0f;
#pragma unroll
  for (int i = 0; i < 4; ++i) {
    float zv = ((const float*)&z)[i];
    pel += zv * al[c + i];
    per += zv * ar[c + i];
  }
  pel = waveSum(pel);
  per = waveSum(per);
  if (lane == 0) { el[wid] = pel; er[wid] = per; }
}

// ---- head: mean(sigmoid(h @ Wlin + blin)); wave per node ----
__global__ void __launch_bounds__(256)
k_final(const float* __restrict__ H, const float* __restrict__ Wlin,
        const float* __restrict__ blin, float* __restrict__ out) {
  int wid  = (blockIdx.x * blockDim.x + threadIdx.x) >> 5;
  int lane = threadIdx.x & 31;
  if (wid >= N_NODES) return;
  int c = lane * 4;
  float4 h = *(const float4*)&H[wid * HID + c];
  float d0 = 0.0f, d1 = 0.0f, d2 = 0.0f;
#pragma unroll
  for (int i = 0; i < 4; ++i) {
    float hv = ((const float*)&h)[i];
    int j = c + i;
    d0 += hv * Wlin[j * 3 + 0];
    d1 += hv * Wlin[j * 3 + 1];
    d2 += hv * Wlin[j * 3 + 2];
  }
  d0 = waveSum(d0); d1 = waveSum(d1); d2 = waveSum(d2);
  if (lane == 0) {
    float s0 = 1.0f / (1.0f + expf(-(d0 + blin[0])));
    float s1 = 1.0f / (1.0f + expf(-(d1 + blin[1])));
    float s2 = 1.0f / (1.0f + expf(-(d2 + blin[2])));
    out[wid] = (s0 + s1 + s2) * (1.0f / 3.0f);
  }
}

extern "C" void kernel_launch(void* const* d_in, const int* in_sizes, int n_in,
                              void* d_out, int out_size, void* d_ws, size_t ws_size,
                              hipStream_t stream) {
  (void)in_sizes; (void)n_in; (void)out_size; (void)ws_size;
  const float* x    = (const float*)d_in[0];
  const int*   src  = (const int*)  d_in[1];
  const int*   dst  = (const int*)  d_in[2];
  const float* W1   = (const float*)d_in[3];
  const float* al1  = (const float*)d_in[4];
  const float* ar1  = (const float*)d_in[5];
  const float* b1   = (const float*)d_in[6];
  const float* W2   = (const float*)d_in[7];
  const float* al2  = (const float*)d_in[8];
  const float* ar2  = (const float*)d_in[9];
  const float* b2   = (const float*)d_in[10];
  const float* Wlin = (const float*)d_in[11];
  const float* blin = (const float*)d_in[12];
  // d_in[13..16] (alpha0/beta0/gamma0/dt) are unused by the reference output.
  float* out = (float*)d_out;

  // Workspace layout (floats): Z | H | el | er | m | denom | ex  (~55 MB)
  float* Z     = (float*)d_ws;
  float* H     = Z + (size_t)N_NODES * HID;
  float* el    = H + (size_t)N_NODES * HID;
  float* er    = el + N_NODES;
  float* m     = er + N_NODES;
  float* denom = m + N_NODES;
  float* ex    = denom + N_NODES;

  const int nodeBlocks = (N_NODES + 7) / 8;        // wave-per-node kernels
  const int edgeBlocks = (N_EDGES + 255) / 256;    // thread-per-edge kernels
  const int aggBlocks  = (N_EDGES + 7) / 8;        // wave-per-edge kernel
  const int elemBlocks = (N_NODES * HID + 255) / 256;
  const int gemmBlocks = N_NODES / 16;             // 3125, exact

  // ----- Layer 1 -----
  k_l1_node   <<<nodeBlocks, 256, 0, stream>>>(x, W1, al1, ar1, Z, el, er);
  k_init      <<<elemBlocks, 256, 0, stream>>>(H, m, denom);
  k_edge_logits<<<edgeBlocks, 256, 0, stream>>>(src, dst, el, er, ex, m);
  k_edge_exp  <<<edgeBlocks, 256, 0, stream>>>(dst, m, ex, denom);
  k_aggregate <<<aggBlocks, 256, 0, stream>>>(src, dst, ex, denom, Z, H);
  k_bias_relu <<<elemBlocks, 256, 0, stream>>>(H, b1);

  // ----- Layer 2 -----
  k_gemm_wmma <<<gemmBlocks, 256, 0, stream>>>(H, W2, Z);     // Z = h1 @ W2
  k_node_attn <<<nodeBlocks, 256, 0, stream>>>(Z, al2, ar2, el, er);
  k_init      <<<elemBlocks, 256, 0, stream>>>(H, m, denom);  // reuse H for h2
  k_edge_logits<<<edgeBlocks, 256, 0, stream>>>(src, dst, el, er, ex, m);
  k_edge_exp  <<<edgeBlocks, 256, 0, stream>>>(dst, m, ex, denom);
  k_aggregate <<<aggBlocks, 256, 0, stream>>>(src, dst, ex, denom, Z, H);
  k_bias_relu <<<elemBlocks, 256, 0, stream>>>(H, b2);

  // ----- Head -----
  k_final     <<<nodeBlocks, 256, 0, stream>>>(H, Wlin, blin, out);
}